// ImageGenerator_78245714198954
// MI455X (gfx1250) — compile-verified
//
#include <hip/hip_runtime.h>
#include <math.h>

#ifndef M_PI
#define M_PI 3.14159265358979323846
#endif

#define BLOCK 256
#define TILE  1024   // floats per O-array tile in LDS (3 arrays -> 12 KB LDS)

// ---------------------------------------------------------------------------
// CDNA5 async global->LDS staging (ASYNCcnt path, gfx1250-specific)
// ---------------------------------------------------------------------------
__device__ __forceinline__ void async_load_f32(const float* gsrc, float* ldst) {
  unsigned loff = (unsigned)(size_t)ldst;               // flat low32 == LDS offset
  unsigned long long gaddr = (unsigned long long)(size_t)gsrc;
  asm volatile("global_load_async_to_lds_b32 %0, %1, off"
               :: "v"(loff), "v"(gaddr) : "memory");
}

__device__ __forceinline__ void async_load_f32x4(const float* gsrc, float* ldst) {
  unsigned loff = (unsigned)(size_t)ldst;
  unsigned long long gaddr = (unsigned long long)(size_t)gsrc;
  asm volatile("global_load_async_to_lds_b128 %0, %1, off"
               :: "v"(loff), "v"(gaddr) : "memory");
}

__device__ __forceinline__ void wait_async0() {
  asm volatile("s_wait_asynccnt 0" ::: "memory");
}

// ---------------------------------------------------------------------------
// One (b,pixel,o) evaluation.
// jnp.sinc(x) = sin(pi x)/(pi x); x = t*qz/(2pi) -> sin(t*qz/2)/(t*qz/2)
// ---------------------------------------------------------------------------
__device__ __forceinline__ float point_term(float dqx, float dqy, float dqz,
                                            float ox, float oy, float oz,
                                            float half_t, float thickness,
                                            float neg_inv_w2) {
  float qx = dqx - ox;
  float qy = dqy - oy;
  float qz = dqz - oz;
  float r2 = fmaf(qy, qy, qx * qx);
  float g  = __expf(r2 * neg_inv_w2);          // v_exp_f32 (WX == WY, fused)
  float arg = half_t * qz;
  float sa  = __sinf(arg);                     // v_sin_f32
  float inv = __builtin_amdgcn_rcpf(arg);      // v_rcp_f32
  float sc  = sa * inv;
  float s2  = sc * sc;
  s2 = (arg * arg < 1e-20f) ? 1.0f : s2;       // sinc(0) = 1
  return thickness * s2 * g;
}

// ---------------------------------------------------------------------------
// Main kernel: grid.x covers outputs (one thread per (b,pixel)),
// grid.y = S o-slices; each block async-stages its O tile into LDS.
// ---------------------------------------------------------------------------
__global__ __launch_bounds__(BLOCK)
void ImageGenerator_78245714198954_main(
    const float* __restrict__ params,
    const float* __restrict__ detQx, const float* __restrict__ detQy,
    const float* __restrict__ detQz,
    const float* __restrict__ OQx, const float* __restrict__ OQy,
    const float* __restrict__ OQz,
    float* __restrict__ dst, int out_n, int npix, int nO, int chunk) {
  __shared__ __align__(16) float sOx[TILE];
  __shared__ __align__(16) float sOy[TILE];
  __shared__ __align__(16) float sOz[TILE];

  const int tid = (int)threadIdx.x;
  const int gid = (int)blockIdx.x * BLOCK + tid;
  const int g   = gid < out_n ? gid : out_n - 1;   // clamp: all threads stage/sync
  const int b   = g / npix;
  const int p   = g - b * npix;

  // Compile-time physics constants (double precision at compile time)
  constexpr double dWL   = 12.398 / 11.3;
  constexpr double dK    = 2.0 * M_PI / dWL;
  constexpr double dPIX  = 5.5e-05 * 2.0;
  constexpr double dDIST = 0.85;
  constexpr double dW    = dK * dPIX / dDIST / 10.0;   // WX == WY
  const float neg_inv_w2 = (float)(-1.0 / (dW * dW));
  const float tpocl      = (float)(2.0 * M_PI / 4.013 * 2.0);  // 2*pi/C*L
  const float d2r        = (float)(M_PI / 180.0);

  const float thickness = params[b * 4 + 0];
  const float strain    = params[b * 4 + 1];
  const float tlr       = params[b * 4 + 2] * d2r;
  const float tud       = params[b * 4 + 3] * d2r;
  const float shift     = tpocl / (1.0f + strain);
  const float half_t    = 0.5f * thickness;

  const float dqx = detQx[p] + shift * tlr;
  const float dqy = detQy[p] + shift * tud;
  const float dqz = detQz[p] - shift;

  // o-slice for this block; chunk is a multiple of 4 -> slice starts are
  // 16B-aligned in the O arrays, enabling b128 async copies.
  const int o0 = (int)blockIdx.y * chunk;
  const int o1 = min(nO, o0 + chunk);

  float acc = 0.0f;

  for (int base = o0; base < o1; base += TILE) {
    const int cnt = min(TILE, o1 - base);

    __syncthreads();  // previous tile fully consumed before overwrite

    // Bulk: 16B-per-lane async copies (512B per wave-issue).
    const int nvec = cnt >> 2;
    for (int v = tid; v < nvec; v += BLOCK) {
      const int i = v << 2;
      async_load_f32x4(OQx + base + i, sOx + i);
      async_load_f32x4(OQy + base + i, sOy + i);
      async_load_f32x4(OQz + base + i, sOz + i);
    }
    // Tail (cnt % 4 elements): 4B async copies.
    for (int i = (nvec << 2) + tid; i < cnt; i += BLOCK) {
      async_load_f32(OQx + base + i, sOx + i);
      async_load_f32(OQy + base + i, sOy + i);
      async_load_f32(OQz + base + i, sOz + i);
    }
    wait_async0();    // this wave's async copies landed in LDS
    __syncthreads();  // every wave's copies landed

    int i = 0;
    for (; i + 4 <= cnt; i += 4) {      // wave-uniform ds_load_b128 broadcasts
      float4 ox = *(const float4*)(sOx + i);
      float4 oy = *(const float4*)(sOy + i);
      float4 oz = *(const float4*)(sOz + i);
      acc += point_term(dqx, dqy, dqz, ox.x, oy.x, oz.x, half_t, thickness, neg_inv_w2);
      acc += point_term(dqx, dqy, dqz, ox.y, oy.y, oz.y, half_t, thickness, neg_inv_w2);
      acc += point_term(dqx, dqy, dqz, ox.z, oy.z, oz.z, half_t, thickness, neg_inv_w2);
      acc += point_term(dqx, dqy, dqz, ox.w, oy.w, oz.w, half_t, thickness, neg_inv_w2);
    }
    for (; i < cnt; ++i)
      acc += point_term(dqx, dqy, dqz, sOx[i], sOy[i], sOz[i], half_t, thickness, neg_inv_w2);
  }

  if (gid < out_n) dst[(size_t)blockIdx.y * out_n + gid] = acc;
}

// Deterministic fixed-order reduction of the S o-slice partials.
__global__ __launch_bounds__(BLOCK)
void ImageGenerator_78245714198954_reduce(const float* __restrict__ partial,
                                          float* __restrict__ out, int n, int S) {
  int i = (int)blockIdx.x * BLOCK + (int)threadIdx.x;
  if (i < n) {
    float a = 0.0f;
    for (int s = 0; s < S; ++s) a += partial[(size_t)s * n + i];
    out[i] = a;
  }
}

extern "C" void kernel_launch(void* const* d_in, const int* in_sizes, int n_in,
                              void* d_out, int out_size, void* d_ws, size_t ws_size,
                              hipStream_t stream) {
  const float* params = (const float*)d_in[0];
  const float* detQx  = (const float*)d_in[1];
  const float* detQy  = (const float*)d_in[2];
  const float* detQz  = (const float*)d_in[3];
  const float* OQx    = (const float*)d_in[4];
  const float* OQy    = (const float*)d_in[5];
  const float* OQz    = (const float*)d_in[6];

  const int npix = in_sizes[1];   // 64*64
  const int nO   = in_sizes[4];   // donut point count

  // Split the o-dimension into S deterministic slices if workspace allows
  // (S depends only on ws_size/out_size -> identical work every call).
  int S = 1;
  const size_t lane = (size_t)out_size * sizeof(float);
  if (ws_size >= 2 * lane) {
    size_t smax = ws_size / lane;
    S = (int)(smax < 8 ? smax : 8);
  }
  // Per-slice chunk, rounded up to a multiple of 4 floats for 16B-aligned
  // b128 async copies.
  int chunk = (nO + S - 1) / S;
  chunk = (chunk + 3) & ~3;

  float* dst = (S > 1) ? (float*)d_ws : (float*)d_out;
  dim3 grid((out_size + BLOCK - 1) / BLOCK, S);
  ImageGenerator_78245714198954_main<<<grid, BLOCK, 0, stream>>>(
      params, detQx, detQy, detQz, OQx, OQy, OQz, dst, out_size, npix, nO, chunk);

  if (S > 1) {
    ImageGenerator_78245714198954_reduce<<<(out_size + BLOCK - 1) / BLOCK, BLOCK, 0, stream>>>(
        (const float*)d_ws, (float*)d_out, out_size, S);
  }
}